// CIntegration_3487513444382
// MI455X (gfx1250) — compile-verified
//
#include <hip/hip_runtime.h>

// Problem constants (match reference)
#define EMB     256
#define NRG     32
#define NSG     32
#define NPC     32
#define NT      96          // NRG + NSG + NPC
#define OUTD    352         // EMB + NT
#define TPB     256         // tokens per block
#define THREADS 256

#define AS1 __attribute__((address_space(1)))
#define AS3 __attribute__((address_space(3)))

__device__ __forceinline__ float onehot_val(int p, int rg, int sg, int pc) {
    // ct layout: [one_hot(rg,32) | one_hot(sg,32) | one_hot(pc,32)]
    const int ref = (p < NRG) ? rg : (p < NRG + NSG) ? (NRG + sg) : (NRG + NSG + pc);
    return (p == ref) ? 1.0f : 0.0f;
}

__global__ __launch_bounds__(THREADS)
void gate_onehot_kernel(const float* __restrict__ vt,
                        const int*   __restrict__ rgap,
                        const int*   __restrict__ sgap,
                        const int*   __restrict__ pcount,
                        const float* __restrict__ Wm,     // [EMB, NT] row-major
                        float*       __restrict__ out,    // [ntok, OUTD]
                        int ntok)
{
    // W transposed into LDS: sW[n*EMB + e] = W[e*NT + n]  (96 KB)
    __shared__ __align__(16) float sW[NT * EMB];
    __shared__ int sRG[TPB];
    __shared__ int sSG[TPB];
    __shared__ int sPC[TPB];

    const int tid = threadIdx.x;
    const int t0  = blockIdx.x * TPB;

    // ---------------- Stage W (transposed) + per-block indices into LDS ----------------
#if __has_builtin(__builtin_amdgcn_global_load_async_to_lds_b32)
    // CDNA5 async-to-LDS data path (tracked with ASYNCcnt)
    for (int i = tid; i < NT * EMB; i += THREADS) {
        const int e = i / NT;
        const int n = i - e * NT;
        __builtin_amdgcn_global_load_async_to_lds_b32(
            (AS1 int*)(void*)(Wm + i),            // coalesced global side
            (AS3 int*)(void*)(&sW[n * EMB + e]),  // transposing LDS scatter
            0, 0);
    }
    for (int i = tid; i < TPB; i += THREADS) {
        const int t = t0 + i;
        if (t < ntok) {
            __builtin_amdgcn_global_load_async_to_lds_b32(
                (AS1 int*)(void*)(rgap + t),   (AS3 int*)(void*)(&sRG[i]), 0, 0);
            __builtin_amdgcn_global_load_async_to_lds_b32(
                (AS1 int*)(void*)(sgap + t),   (AS3 int*)(void*)(&sSG[i]), 0, 0);
            __builtin_amdgcn_global_load_async_to_lds_b32(
                (AS1 int*)(void*)(pcount + t), (AS3 int*)(void*)(&sPC[i]), 0, 0);
        }
    }
#if __has_builtin(__builtin_amdgcn_s_wait_asynccnt)
    __builtin_amdgcn_s_wait_asynccnt(0);
#else
    asm volatile("s_wait_asynccnt 0" ::: "memory");
#endif
#else
#warning "gfx1250 async-to-LDS builtin NOT available on this toolchain; using VALU staging fallback"
    for (int i = tid; i < NT * EMB; i += THREADS) {
        const int e = i / NT;
        const int n = i - e * NT;
        sW[n * EMB + e] = Wm[i];
    }
    for (int i = tid; i < TPB; i += THREADS) {
        const int t = t0 + i;
        sRG[i] = (t < ntok) ? rgap[t]   : 0;
        sSG[i] = (t < ntok) ? sgap[t]   : 0;
        sPC[i] = (t < ntok) ? pcount[t] : 0;
    }
#endif
    __syncthreads();

    // ---------------- Streaming compute: 4 tokens per iteration ----------------
    const int lane4 = (tid & 63) * 4;   // element offset within the 256-wide theta
    const int sub   = tid >> 6;         // which of the 4 tokens this thread serves

    #pragma unroll 2
    for (int tb = 0; tb < TPB; tb += 4) {
        const int t = t0 + tb + sub;
        if (t < ntok) {
            const int li = tb + sub;
            const int rg = sRG[li];
            const int sg = sSG[li];
            const int pc = sPC[li];

            // Prefetch next iteration's vt stream (global_prefetch_b8)
            if (t + 4 < ntok)
                __builtin_prefetch(vt + (size_t)(t + 4) * EMB + lane4, 0, 3);

            const float4 v  = *(const float4*)(vt + (size_t)t * EMB + lane4);
            const float4 w0 = *(const float4*)(&sW[(rg)             * EMB + lane4]);
            const float4 w1 = *(const float4*)(&sW[(NRG + sg)       * EMB + lane4]);
            const float4 w2 = *(const float4*)(&sW[(NRG + NSG + pc) * EMB + lane4]);

            float4 th;
            th.x = v.x * (w0.x + w1.x + w2.x);
            th.y = v.y * (w0.y + w1.y + w2.y);
            th.z = v.z * (w0.z + w1.z + w2.z);
            th.w = v.w * (w0.w + w1.w + w2.w);

            *(float4*)(out + (size_t)t * OUTD + lane4) = th;
        }

        // 96 one-hot floats per token, 4 tokens -> threads 0..95, float4 each
        if (tid < 96) {
            const int tt = tid / 24;            // token 0..3 of this group
            const int p  = (tid % 24) * 4;      // one-hot position 0..92 (stays in one segment)
            const int t2 = t0 + tb + tt;
            if (t2 < ntok) {
                const int li2 = tb + tt;
                const int rg = sRG[li2];
                const int sg = sSG[li2];
                const int pc = sPC[li2];
                float4 c;
                c.x = onehot_val(p + 0, rg, sg, pc);
                c.y = onehot_val(p + 1, rg, sg, pc);
                c.z = onehot_val(p + 2, rg, sg, pc);
                c.w = onehot_val(p + 3, rg, sg, pc);
                *(float4*)(out + (size_t)t2 * OUTD + EMB + p) = c;
            }
        }
    }
}

extern "C" void kernel_launch(void* const* d_in, const int* in_sizes, int n_in,
                              void* d_out, int out_size, void* d_ws, size_t ws_size,
                              hipStream_t stream) {
    const float* vt     = (const float*)d_in[0];
    const int*   rgap   = (const int*)  d_in[1];
    const int*   sgap   = (const int*)  d_in[2];
    const int*   pcount = (const int*)  d_in[3];
    const float* W      = (const float*)d_in[4];
    float*       out    = (float*)d_out;

    const int ntok    = in_sizes[1];                 // B*S (rgap flat count)
    const int nblocks = (ntok + TPB - 1) / TPB;

    hipLaunchKernelGGL(gate_onehot_kernel, dim3(nblocks), dim3(THREADS), 0, stream,
                       vt, rgap, sgap, pcount, W, out, ntok);
}